// TimeVaryingLinearFlow_44650480009359
// MI455X (gfx1250) — compile-verified
//
#include <hip/hip_runtime.h>
#include <hip/hip_bf16.h>
#include <math.h>

typedef __attribute__((ext_vector_type(2))) float v2f;
typedef __attribute__((ext_vector_type(4))) float v4f;
typedef __attribute__((ext_vector_type(8))) float v8f;

#define T_STEPS 2048
#define NPART   8192
#define HID     128

// ---------------------------------------------------------------------------
// Layer 1: h1[t][h] = tanh(t_avg[t] * W1[h] + b1[h]);  also emit dt[t].
// ---------------------------------------------------------------------------
__global__ __launch_bounds__(256) void k_layer1(const float* __restrict__ t,
                                                const float* __restrict__ W1,
                                                const float* __restrict__ b1,
                                                float* __restrict__ h1,
                                                float* __restrict__ dt) {
  int idx = blockIdx.x * blockDim.x + threadIdx.x;   // T*H threads
  int ti = idx >> 7;                                 // / 128
  int h  = idx & 127;
  float tc = t[ti];
  float tp = (ti > 0) ? t[ti - 1] : 0.0f;
  if (h == 0) dt[ti] = tc - tp;
  float tavg = 0.5f * (tc + tp);
  h1[idx] = tanhf(fmaf(tavg, W1[h], b1[h]));
}

// ---------------------------------------------------------------------------
// Layer 2: h2 = tanh(h1 @ W2 + b2)   (2048x128)@(128x128), WMMA f32 16x16x4.
// One wave per 16x16 output tile, K-loop of 32 chained WMMAs.
// A operand (16x4 f32): lane M = lane%16, K sub-offset {0,1}/{2,3} per half.
// B operand (4x16 f32): lane N = lane%16, K striped across the 2 VGPRs with
// lane halves covering K={0,1} / K={2,3}.
// ---------------------------------------------------------------------------
__global__ __launch_bounds__(256) void k_layer2(const float* __restrict__ h1,
                                                const float* __restrict__ W2,
                                                const float* __restrict__ b2,
                                                float* __restrict__ h2) {
  int gid  = blockIdx.x * blockDim.x + threadIdx.x;
  int w    = gid >> 5;
  int lane = gid & 31;
  int t0 = (w >> 3) << 4;          // 128 row-tiles of 16 timesteps
  int n0 = (w & 7) << 4;           // 8 col-tiles of 16 outputs
  int col = lane & 15;
  int kb  = (lane < 16) ? 0 : 2;   // K sub-offset for this lane half

  const float* arow = h1 + (t0 + col) * HID + kb;   // A row for M = col
  v8f acc = {};
  for (int k0 = 0; k0 < HID; k0 += 4) {
    v2f a = *(const v2f*)(arow + k0);
    v2f b;
    b.x = W2[(k0 + kb) * HID + n0 + col];
    b.y = W2[(k0 + kb + 1) * HID + n0 + col];
    acc = __builtin_amdgcn_wmma_f32_16x16x4_f32(false, a, false, b,
                                                (short)0, acc, false, false);
  }
  float bn = b2[n0 + col];
  int mofs = (lane < 16) ? 0 : 8;  // C/D: VGPR r holds M=r (low half), M=r+8
  #pragma unroll
  for (int r = 0; r < 8; ++r) {
    int trow = t0 + r + mofs;
    h2[trow * HID + n0 + col] = tanhf(acc[r] + bn);
  }
}

// ---------------------------------------------------------------------------
// Layer 3: A[t] = ((h2 @ W3 + b3) + M0) * dt[t]   (2048x128)@(128x16).
// ---------------------------------------------------------------------------
__global__ __launch_bounds__(256) void k_layer3(const float* __restrict__ h2,
                                                const float* __restrict__ W3,
                                                const float* __restrict__ b3,
                                                const float* __restrict__ M0,
                                                const float* __restrict__ dt,
                                                float* __restrict__ A) {
  int gid  = blockIdx.x * blockDim.x + threadIdx.x;
  int w    = gid >> 5;
  int lane = gid & 31;
  int t0  = w << 4;                // 128 row-tiles, single 16-wide col tile
  int col = lane & 15;
  int kb  = (lane < 16) ? 0 : 2;

  const float* arow = h2 + (t0 + col) * HID + kb;
  v8f acc = {};
  for (int k0 = 0; k0 < HID; k0 += 4) {
    v2f a = *(const v2f*)(arow + k0);
    v2f b;
    b.x = W3[(k0 + kb) * 16 + col];
    b.y = W3[(k0 + kb + 1) * 16 + col];
    acc = __builtin_amdgcn_wmma_f32_16x16x4_f32(false, a, false, b,
                                                (short)0, acc, false, false);
  }
  float bias = b3[col] + M0[col];
  int mofs = (lane < 16) ? 0 : 8;
  #pragma unroll
  for (int r = 0; r < 8; ++r) {
    int trow = t0 + r + mofs;
    A[trow * 16 + col] = (acc[r] + bias) * dt[trow];
  }
}

// ---------------------------------------------------------------------------
// Prefix-sum over time of the 16 matrix elements. 256 threads = 16 elements
// x 16 chunks of 128 steps; chunk sums exchanged through LDS.
// ---------------------------------------------------------------------------
__global__ __launch_bounds__(256) void k_cumsum(const float* __restrict__ A,
                                                float* __restrict__ Asum) {
  __shared__ float csum[16][17];
  int e = threadIdx.x & 15;      // matrix element
  int c = threadIdx.x >> 4;      // chunk
  const int CHUNK = T_STEPS / 16;
  int base = c * CHUNK;
  float s = 0.0f;
  for (int i = 0; i < CHUNK; ++i) s += A[(base + i) * 16 + e];
  csum[c][e] = s;
  __syncthreads();
  float acc = 0.0f;
  for (int cc = 0; cc < c; ++cc) acc += csum[cc][e];
  for (int i = 0; i < CHUNK; ++i) {
    acc += A[(base + i) * 16 + e];
    Asum[(base + i) * 16 + e] = acc;
  }
}

// ---------------------------------------------------------------------------
// 4x4 matrix exponential, one thread per timestep: scaling & squaring with a
// 16-term Horner Taylor series (argument norm scaled to <= 0.5).
// ---------------------------------------------------------------------------
__device__ __forceinline__ void mm4(const float* a, const float* b, float* c) {
  #pragma unroll
  for (int i = 0; i < 4; ++i)
    #pragma unroll
    for (int j = 0; j < 4; ++j) {
      float s = 0.0f;
      #pragma unroll
      for (int k = 0; k < 4; ++k) s = fmaf(a[i * 4 + k], b[k * 4 + j], s);
      c[i * 4 + j] = s;
    }
}

__global__ __launch_bounds__(256) void k_expm(const float* __restrict__ Asum,
                                              float* __restrict__ E) {
  int t = blockIdx.x * blockDim.x + threadIdx.x;
  float M[16];
  #pragma unroll
  for (int i = 0; i < 4; ++i) {
    v4f v = *(const v4f*)(Asum + t * 16 + i * 4);
    M[i * 4 + 0] = v.x; M[i * 4 + 1] = v.y;
    M[i * 4 + 2] = v.z; M[i * 4 + 3] = v.w;
  }
  float nrm = 0.0f;
  #pragma unroll
  for (int i = 0; i < 4; ++i) {
    float s = fabsf(M[i * 4]) + fabsf(M[i * 4 + 1]) +
              fabsf(M[i * 4 + 2]) + fabsf(M[i * 4 + 3]);
    nrm = fmaxf(nrm, s);
  }
  int j = 0;
  if (nrm > 0.5f) {
    j = (int)ceilf(log2f(nrm * 2.0f));
    j = (j < 0) ? 0 : ((j > 46) ? 46 : j);
  }
  float sc = exp2f(-(float)j);
  #pragma unroll
  for (int e = 0; e < 16; ++e) M[e] *= sc;

  float Em[16];
  #pragma unroll
  for (int e = 0; e < 16; ++e) Em[e] = ((e & 3) == (e >> 2)) ? 1.0f : 0.0f;
  for (int k = 16; k >= 1; --k) {            // Em = I + M*Em/k
    float Tm[16];
    mm4(M, Em, Tm);
    float inv = 1.0f / (float)k;
    #pragma unroll
    for (int e = 0; e < 16; ++e)
      Em[e] = (((e & 3) == (e >> 2)) ? 1.0f : 0.0f) + Tm[e] * inv;
  }
  for (int s2 = 0; s2 < j; ++s2) {           // repeated squaring
    float Tm[16];
    mm4(Em, Em, Tm);
    #pragma unroll
    for (int e = 0; e < 16; ++e) Em[e] = Tm[e];
  }
  #pragma unroll
  for (int i = 0; i < 4; ++i) {
    v4f v; v.x = Em[i * 4]; v.y = Em[i * 4 + 1];
    v.z = Em[i * 4 + 2]; v.w = Em[i * 4 + 3];
    *(v4f*)(E + t * 16 + i * 4) = v;
  }
}

// ---------------------------------------------------------------------------
// Streaming apply: out[t][n][i] = sum_j E[t][i][j] * x[n][j].
// One WMMA per (16 particles x 4 timesteps): A = x tile (loaded ONCE per
// wave), B packs 4 E-matrices as columns c = t_sub*4 + i. The hot loop is one
// 8B E load + one v_wmma_f32_16x16x4_f32 + 8 stores -> store-bandwidth bound
// (256 MB of output at 23.3 TB/s ~= 11 us floor). E (128 KB) stays in L2.
// ---------------------------------------------------------------------------
#define TCHUNKS 8

__global__ __launch_bounds__(256) void k_apply(const float* __restrict__ x,
                                               const float* __restrict__ E,
                                               float* __restrict__ out) {
  int gid  = blockIdx.x * blockDim.x + threadIdx.x;
  int w    = gid >> 5;
  int lane = gid & 31;
  int nb = w & (NPART / 16 - 1);       // particle block 0..511
  int tc = w >> 9;                     // timestep chunk 0..TCHUNKS-1
  int col = lane & 15;
  int jb  = (lane < 16) ? 0 : 2;       // K (j) sub-offset per lane half

  // A operand: rows = 16 particles, cols = 4 dims (reused for all t)
  v2f a = *(const v2f*)(x + (nb * 16 + col) * 4 + jb);

  int tsub = col >> 2;                 // column -> (timestep-in-group, i)
  int i    = col & 3;
  int mofs = (lane < 16) ? 0 : 8;
  const int TG_PER = (T_STEPS / 4) / TCHUNKS;   // 64 groups per wave
  int tg0 = tc * TG_PER;

  for (int g = 0; g < TG_PER; ++g) {
    int t0 = (tg0 + g) << 2;
    v2f b = *(const v2f*)(E + (t0 + tsub) * 16 + i * 4 + jb);
    v8f acc = {};
    acc = __builtin_amdgcn_wmma_f32_16x16x4_f32(false, a, false, b,
                                                (short)0, acc, false, false);
    int trow = t0 + tsub;
    float* op = out + ((size_t)trow * NPART + (size_t)nb * 16) * 4 + i;
    #pragma unroll
    for (int r = 0; r < 8; ++r) {
      op[(size_t)(r + mofs) * 4] = acc[r];   // wave covers the full 256B span
    }
  }
}

// ---------------------------------------------------------------------------
extern "C" void kernel_launch(void* const* d_in, const int* in_sizes, int n_in,
                              void* d_out, int out_size, void* d_ws, size_t ws_size,
                              hipStream_t stream) {
  const float* x  = (const float*)d_in[0];
  const float* t  = (const float*)d_in[1];
  const float* M0 = (const float*)d_in[2];
  const float* W1 = (const float*)d_in[3];
  const float* b1 = (const float*)d_in[4];
  const float* W2 = (const float*)d_in[5];
  const float* b2 = (const float*)d_in[6];
  const float* W3 = (const float*)d_in[7];
  const float* b3 = (const float*)d_in[8];
  float* out = (float*)d_out;

  float* ws   = (float*)d_ws;              // ~2.5 MB total
  float* dt   = ws;                        // T
  float* h1   = dt + T_STEPS;              // T*128
  float* h2   = h1 + T_STEPS * HID;        // T*128
  float* Amat = h2 + T_STEPS * HID;        // T*16
  float* Asum = Amat + T_STEPS * 16;       // T*16
  float* Emat = Asum + T_STEPS * 16;       // T*16

  k_layer1<<<(T_STEPS * HID) / 256, 256, 0, stream>>>(t, W1, b1, h1, dt);
  k_layer2<<<((T_STEPS / 16) * (HID / 16) * 32) / 256, 256, 0, stream>>>(h1, W2, b2, h2);
  k_layer3<<<((T_STEPS / 16) * 32) / 256, 256, 0, stream>>>(h2, W3, b3, M0, dt, Amat);
  k_cumsum<<<1, 256, 0, stream>>>(Amat, Asum);
  k_expm<<<T_STEPS / 256, 256, 0, stream>>>(Asum, Emat);

  int waves = (NPART / 16) * TCHUNKS;      // 4096 waves
  k_apply<<<(waves * 32) / 256, 256, 0, stream>>>(x, Emat, out);
}